// HyperConnections_22608707846863
// MI455X (gfx1250) — compile-verified
//
#include <hip/hip_runtime.h>
#include <cstdint>

#ifndef __has_builtin
#define __has_builtin(x) 0
#endif

typedef __attribute__((ext_vector_type(2))) float v2f;
typedef __attribute__((ext_vector_type(8))) float v8f;
typedef __attribute__((ext_vector_type(4))) int   v4i;

#define DIMD   2048
#define PITCH  2052        // DIM + 4-float pad -> conflict-free WMMA A feeds from LDS
#define NROWS  16          // 4 positions * rate(4) rows = one WMMA M-tile
#define BLOCK  256         // 8 wave32 waves
#define NWAVE  8
#define KSTEP  (DIMD / NWAVE)
#define DSCALE 0.01f
#define EPSV   1e-5f
#define MGROWS 7           // 5 alpha cols + 1 beta col + 1 zero row

// ---------- CDNA5 async global->LDS copy (ASYNCcnt path) ----------
__device__ __forceinline__ void async_copy_b128(const float* gsrc, float* ldst) {
#if __has_builtin(__builtin_amdgcn_global_load_async_to_lds_b128)
  __builtin_amdgcn_global_load_async_to_lds_b128(
      (__attribute__((address_space(1))) v4i*)gsrc,
      (__attribute__((address_space(3))) v4i*)ldst,
      0, 0);
#else
  asm volatile("global_load_async_to_lds_b128 %0, %1, off"
               :: "v"((unsigned)(uintptr_t)ldst), "v"(gsrc)
               : "memory");
#endif
}

__device__ __forceinline__ void wait_async0() {
#if __has_builtin(__builtin_amdgcn_s_wait_asynccnt)
  __builtin_amdgcn_s_wait_asynccnt(0);
#else
  asm volatile("s_wait_asynccnt 0x0" ::: "memory");
#endif
}

// ---------- Kernel 0: Mgt[k][d] = gamma[d]*w_k[d] (k=0..4 alpha, 5 beta, 6 zeros)
//            plus fold constants G[k] = sum gamma*w_k, Bc[k] = sum ln_beta*w_k ----------
__global__ __launch_bounds__(256) void hc_precompute(
    const float* __restrict__ gamma, const float* __restrict__ lnb,
    const float* __restrict__ Wa,    const float* __restrict__ Wb,
    float* __restrict__ Mgt, float* __restrict__ Gc, float* __restrict__ Bco)
{
  __shared__ float red[256];
  const int tid = threadIdx.x;
  float pg[6] = {0.f,0.f,0.f,0.f,0.f,0.f};
  float pb[6] = {0.f,0.f,0.f,0.f,0.f,0.f};
  for (int d = tid; d < DIMD; d += 256) {
    const float g = gamma[d], lb = lnb[d];
    float w[6];
#pragma unroll
    for (int k = 0; k < 5; ++k) w[k] = Wa[d * 5 + k];
    w[5] = Wb[d];
#pragma unroll
    for (int k = 0; k < 6; ++k) {
      const float gw = g * w[k];
      Mgt[k * DIMD + d] = gw;          // transposed: per-column contiguous in d
      pg[k] += gw;
      pb[k] = fmaf(lb, w[k], pb[k]);
    }
    Mgt[6 * DIMD + d] = 0.f;           // zero row for lanes m >= 6
  }
  for (int qi = 0; qi < 12; ++qi) {
    red[tid] = (qi < 6) ? pg[qi] : pb[qi - 6];
    __syncthreads();
    for (int off = 128; off > 0; off >>= 1) {
      if (tid < off) red[tid] += red[tid + off];
      __syncthreads();
    }
    if (tid == 0) { if (qi < 6) Gc[qi] = red[0]; else Bco[qi - 6] = red[0]; }
    __syncthreads();
  }
}

// ---------- Fused kernel: one WG = 4 (b,l) positions = 16 rows of h ----------
__global__ __launch_bounds__(BLOCK) void hc_fused(
    const float* __restrict__ h,  const float* __restrict__ h_o,
    const float* __restrict__ Mgt, const float* __restrict__ Gc,
    const float* __restrict__ Bco,
    const float* __restrict__ salpha,  // (4,5)
    const float* __restrict__ sbeta,   // (4,)
    float* __restrict__ out)
{
  extern __shared__ float smem[];
  float* hs      = smem;                         // NROWS * PITCH
  float* Spart   = smem + NROWS * PITCH;         // NWAVE * 16 * 8
  float* sumpart = Spart + NWAVE * 128;          // NWAVE * 32
  float* sqpart  = sumpart + NWAVE * 32;         // NWAVE * 32
  float* mixc    = sqpart + NWAVE * 32;          // 4 pos * 4 n * 4 j
  float* betc    = mixc + 64;                    // 16

  const int tid  = threadIdx.x;
  const int lane = tid & 31;
  const int wv   = tid >> 5;
  const int wg   = blockIdx.x;
  const long long row0 = (long long)wg * NROWS;
  const float* htile = h + row0 * DIMD;

  // ---- Stage 16 x 2048 f32 tile of h into LDS with async B128 copies ----
#pragma unroll
  for (int i = 0; i < 32; ++i) {
    const int flat = i * BLOCK + tid;   // float4 index 0..8191
    const int row  = flat >> 9;         // / 512
    const int c4   = flat & 511;
    async_copy_b128(htile + row * DIMD + c4 * 4, hs + row * PITCH + c4 * 4);
  }
  wait_async0();
  __syncthreads();

  // ---- Phase 1: S = h_tile(16xK) @ M(Kx16) via V_WMMA_F32_16X16X4_F32,
  //      K split across the 8 waves; per-row sum/sumsq ride along in VALU.
  //      B feed is branchless: lanes m>=6 stream the zero row of Mgt. ----
  const int half = lane >> 4;
  const int m    = lane & 15;           // A row index / B,C column index
  const int koff = half * 2;            // lanes 0-15: K0,K1; lanes 16-31: K2,K3
  const int kbase = wv * KSTEP;
  const int mclamp = (m < 6) ? m : 6;
  const float* aptr = hs  + m * PITCH + kbase + koff;        // LDS, b64 per step
  const float* bptr = Mgt + mclamp * DIMD + kbase + koff;    // global, b64 per step

  v8f acc = {0.f,0.f,0.f,0.f,0.f,0.f,0.f,0.f};
  float s = 0.f, q = 0.f;
  for (int it = 0; it < KSTEP / 4; ++it) {
    v2f a = *(const v2f*)(aptr + it * 4);   // ds_load_b64, conflict-free (pad 4)
    v2f b = *(const v2f*)(bptr + it * 4);   // global_load_b64, L0/L2-resident
    acc = __builtin_amdgcn_wmma_f32_16x16x4_f32(false, a, false, b,
                                                (short)0, acc, false, false);
    s += a.x + a.y;
    q = fmaf(a.x, a.x, fmaf(a.y, a.y, q));
  }
  sumpart[wv * 32 + lane] = s;
  sqpart [wv * 32 + lane] = q;
#pragma unroll
  for (int v = 0; v < 8; ++v) {
    if (m < 6) Spart[wv * 128 + (v + 8 * half) * 8 + m] = acc[v];  // C: row v+8*half, col m
  }
  __syncthreads();

  // ---- Finalize coefficients (deterministic cross-wave combine, 16 threads) ----
  if (tid < 16) {
    const int r = tid, nn = r & 3, p = r >> 2;
    float sm = 0.f, sq = 0.f;
#pragma unroll
    for (int w2 = 0; w2 < NWAVE; ++w2) {
      sm += sumpart[w2 * 32 + r] + sumpart[w2 * 32 + 16 + r];
      sq += sqpart [w2 * 32 + r] + sqpart [w2 * 32 + 16 + r];
    }
    const float rN = 1.f / (float)DIMD;
    const float mu  = sm * rN;
    const float var = sq * rN - mu * mu;
    const float inv = rsqrtf(var + EPSV);
#pragma unroll
    for (int k = 1; k <= 5; ++k) {      // alpha cols 1..4, k==5 -> beta column
      float S = 0.f;
#pragma unroll
      for (int w2 = 0; w2 < NWAVE; ++w2) S += Spart[w2 * 128 + r * 8 + k];
      const float dyn = inv * (S - mu * Gc[k]) + Bco[k];
      if (k < 5) mixc[(p * 4 + nn) * 4 + (k - 1)] = DSCALE * dyn + salpha[nn * 5 + k];
      else       betc[r] = DSCALE * dyn + sbeta[nn];
    }
  }
  __syncthreads();

  // ---- Phase 2: out[p][j][:] = beta*h_o + sum_n alpha[n][j+1]*h[n][:] ----
  const long long pos0 = (long long)wg * 4;
#pragma unroll
  for (int i = 0; i < 8; ++i) {
    const int flat = i * BLOCK + tid;   // 0..2047 float4 chunks
    const int p  = flat >> 9;
    const int c4 = flat & 511;
    const int d  = c4 * 4;
    const float4 ho = *(const float4*)(h_o + (pos0 + p) * DIMD + d);
    const float* hp = hs + (p * 4) * PITCH + d;
    const float4 h0 = *(const float4*)(hp);
    const float4 h1 = *(const float4*)(hp + PITCH);
    const float4 h2 = *(const float4*)(hp + 2 * PITCH);
    const float4 h3 = *(const float4*)(hp + 3 * PITCH);
    const float* mc = mixc + p * 16;
    float* op = out + ((pos0 + p) * 4) * DIMD + d;
#pragma unroll
    for (int j = 0; j < 4; ++j) {
      const float bb = betc[p * 4 + j];
      const float c0 = mc[0 * 4 + j], c1 = mc[1 * 4 + j];
      const float c2 = mc[2 * 4 + j], c3 = mc[3 * 4 + j];
      float4 o;
      o.x = fmaf(ho.x, bb, fmaf(c0, h0.x, fmaf(c1, h1.x, fmaf(c2, h2.x, c3 * h3.x))));
      o.y = fmaf(ho.y, bb, fmaf(c0, h0.y, fmaf(c1, h1.y, fmaf(c2, h2.y, c3 * h3.y))));
      o.z = fmaf(ho.z, bb, fmaf(c0, h0.z, fmaf(c1, h1.z, fmaf(c2, h2.z, c3 * h3.z))));
      o.w = fmaf(ho.w, bb, fmaf(c0, h0.w, fmaf(c1, h1.w, fmaf(c2, h2.w, c3 * h3.w))));
      *(float4*)(op + j * DIMD) = o;
    }
  }
}

extern "C" void kernel_launch(void* const* d_in, const int* in_sizes, int n_in,
                              void* d_out, int out_size, void* d_ws, size_t ws_size,
                              hipStream_t stream) {
  (void)n_in; (void)out_size; (void)ws_size;
  const float* h   = (const float*)d_in[0];
  const float* ho  = (const float*)d_in[1];
  const float* gam = (const float*)d_in[2];
  const float* lnb = (const float*)d_in[3];
  const float* Wa  = (const float*)d_in[4];
  const float* Wb  = (const float*)d_in[5];
  const float* sa  = (const float*)d_in[6];
  const float* sb  = (const float*)d_in[7];
  float* out = (float*)d_out;
  float* ws  = (float*)d_ws;
  float* Mgt = ws;                         // 7*2048 floats (56 KB), transposed
  float* Gc  = ws + MGROWS * DIMD;         // 6 floats (padded to 8)
  float* Bco = ws + MGROWS * DIMD + 8;     // 6 floats

  const int BL = in_sizes[1] / DIMD;  // number of (b,l) positions = 8192

  hc_precompute<<<1, 256, 0, stream>>>(gam, lnb, Wa, Wb, Mgt, Gc, Bco);

  const size_t smem_bytes =
      (size_t)(NROWS * PITCH + NWAVE * 128 + NWAVE * 32 * 2 + 64 + 16) * sizeof(float);
  hc_fused<<<BL / 4, BLOCK, smem_bytes, stream>>>(h, ho, Mgt, Gc, Bco, sa, sb, out);
}